// VarianceAdaptor_57423712747751
// MI455X (gfx1250) — compile-verified
//
#include <hip/hip_runtime.h>
#include <cstddef>
#include <cstdint>

typedef float v2f __attribute__((ext_vector_type(2)));
typedef float v8f __attribute__((ext_vector_type(8)));
typedef unsigned int v4u __attribute__((ext_vector_type(4)));
typedef int v4i __attribute__((ext_vector_type(4)));
typedef int v8i __attribute__((ext_vector_type(8)));

#define B_      32
#define T_      1024
#define D_      384
#define MAXOUT_ 4096
#define SXPITCH 385   // LDS row pitch in floats (odd -> conflict-free row access)

// =====================================================================
// conv1d(K=3, 384->384, SAME) + bias + relu + layernorm, fp32 WMMA
// X tile staged into LDS by the Tensor Data Mover (iterate mode keeps the
// 385-float padded row pitch).  grid: 2048 blocks, block: 256 threads.
// =====================================================================
__global__ __launch_bounds__(256)
void va_conv_ln_wmma(const float* __restrict__ X,    // (B,T,384)
                     const float* __restrict__ W,    // (3,384,384)
                     const float* __restrict__ bias, // (384)
                     const float* __restrict__ lns,  // (384)
                     const float* __restrict__ lnb,  // (384)
                     float* __restrict__ Y)          // (B,T,384)
{
  __shared__ float sX[18][SXPITCH];   // halo rows t0-1 .. t0+16
  __shared__ float sH[16][SXPITCH];   // relu(conv+bias) staging for layernorm

  const int mt = blockIdx.x;              // 2048 M tiles, 64 per batch
  const int b  = mt >> 6;
  const int t0 = (mt & 63) << 4;
  const float* xb = X + (size_t)b * T_ * D_;
  const int tid  = threadIdx.x;
  const int wave = tid >> 5, lane = tid & 31;
  const int g = lane >> 4;       // half-wave group (K split)
  const int i = lane & 15;       // M index for A, N index for B/C/D

  // SAME padding: zero halo rows the DMA does not cover (batch edges)
  if (t0 == 0)        for (int c = tid; c < D_; c += 256) sX[0][c]  = 0.0f;
  if (t0 == T_ - 16)  for (int c = tid; c < D_; c += 256) sX[17][c] = 0.0f;

  // TDM: one wave issues the async tile DMA (rows clamped to the batch),
  // iterate mode = one 384-elem row per iteration, LDS stride 385 elems.
  if (wave == 0) {
    const int first   = (t0 == 0) ? 0 : t0 - 1;
    const int last    = (t0 + 16 <= T_ - 1) ? t0 + 16 : T_ - 1;
    const int nrows   = last - first + 1;
    const int ldsrow0 = first - (t0 - 1);            // 0 interior, 1 at t0==0
    const unsigned lds_base =
        (unsigned)(size_t)(&sX[ldsrow0][0]);          // low 32 bits of the flat pointer are the LDS address
    const unsigned long long ga =
        (unsigned long long)(size_t)(xb + (size_t)first * D_);

    v4u g0 = {};
    g0[0] = 1u;                                       // count = 1 valid D#
    g0[1] = lds_base;                                 // lds_addr (bytes)
    g0[2] = (unsigned)(ga & 0xFFFFFFFFu);             // global_addr[31:0]
    g0[3] = (unsigned)((ga >> 32) & 0x01FFFFFFu) | (2u << 30); // [56:32]|type=2

    v8i g1 = {};
    g1[0] = (int)((2u << 16) | (1u << 19));           // data_size=4B, iterate_en
    g1[1] = (int)(384u << 16);                        // tensor_dim0 = 384
    g1[2] = (int)(1u << 16);                          // tensor_dim1 = 1
    g1[3] = (int)(384u << 16);                        // tile_dim0 = 384
    g1[4] = 1;                                        // tile_dim1 = 1
    g1[5] = 384;                                      // tensor_dim0_stride
    v4i g2 = {};
    g2[1] = SXPITCH;                                  // lds_addr_increment
    g2[2] = 384;                                      // global_addr_increment
    g2[3] = (nrows - 1) << 16;                        // iterate_count (0 => 1x)
    v4i g3 = {};
    v8i g4 = {};                                      // aux group (clang-23 form)
    __builtin_amdgcn_tensor_load_to_lds(g0, g1, g2, g3, g4, 0);
    __builtin_amdgcn_s_wait_tensorcnt(0);
  }
  __syncthreads();

  // each wave owns 3 of the 24 N tiles
  for (int nt = wave; nt < 24; nt += 8) {
    const int n0 = nt << 4;
    v8f acc = {};
    for (int s = 0; s < 3; ++s) {          // conv taps: t-1, t, t+1
      const float* Ws = W + (size_t)s * D_ * D_ + n0 + i;
      const float* xr = &sX[i + s][0];     // output row i reads input row i+s-1
      for (int k0 = 0; k0 < D_; k0 += 4) {
        const int k = k0 + 2 * g;          // A/B layout: K = 2*group + {0,1}
        v2f a;  a.x  = xr[k];              a.y  = xr[k + 1];
        v2f bb; bb.x = Ws[(size_t)k * D_]; bb.y = Ws[(size_t)(k + 1) * D_];
        acc = __builtin_amdgcn_wmma_f32_16x16x4_f32(false, a, false, bb,
                                                    (short)0, acc, false, false);
      }
    }
    const float bi = bias[n0 + i];
#pragma unroll
    for (int r = 0; r < 8; ++r) {          // C/D layout: VGPR r -> M = r + 8*g
      float v = acc[r] + bi;
      sH[r + 8 * g][n0 + i] = v > 0.0f ? v : 0.0f;
    }
  }
  __syncthreads();

  // layernorm over H=384 per row; wave handles rows {wave, wave+8}
  for (int m = wave; m < 16; m += 8) {
    float s1 = 0.0f, s2 = 0.0f;
    for (int c = lane; c < D_; c += 32) { float v = sH[m][c]; s1 += v; s2 += v * v; }
#pragma unroll
    for (int off = 16; off; off >>= 1) {
      s1 += __shfl_xor(s1, off, 32);
      s2 += __shfl_xor(s2, off, 32);
    }
    const float mu  = s1 * (1.0f / D_);
    const float var = s2 * (1.0f / D_) - mu * mu;
    const float inv = rsqrtf(var + 1e-5f);
    float* yr = Y + (size_t)(b * T_ + t0 + m) * D_;
    for (int c = lane; c < D_; c += 32)
      yr[c] = (sH[m][c] - mu) * inv * lns[c] + lnb[c];
  }
}

// =====================================================================
// fused projection (H->1) + squared-error loss accumulation
// mode 0: ref = log(dur + 1e-8)   mode 1: ref = tgt_f[row]
// both losses divide by B*T = 32768
// =====================================================================
__global__ __launch_bounds__(256)
void va_proj_loss(const float* __restrict__ Hm, const float* __restrict__ pw,
                  const float* __restrict__ pb, const float* __restrict__ tgt_f,
                  const int* __restrict__ tgt_i, int mode,
                  float* __restrict__ loss_slot)
{
  __shared__ float part[8];
  const int wave = threadIdx.x >> 5, lane = threadIdx.x & 31;
  const int row = blockIdx.x * 8 + wave;
  const float* hr = Hm + (size_t)row * D_;
  float s = 0.0f;
  for (int c = lane; c < D_; c += 32) s += hr[c] * pw[c];
#pragma unroll
  for (int off = 16; off; off >>= 1) s += __shfl_xor(s, off, 32);
  if (lane == 0) {
    float val = s + pb[0];
    float ref = (mode == 0) ? logf((float)tgt_i[row] + 1e-8f) : tgt_f[row];
    float d = val - ref;
    part[wave] = d * d * (1.0f / 32768.0f);
  }
  __syncthreads();
  if (threadIdx.x == 0) {
    float acc = 0.0f;
#pragma unroll
    for (int w = 0; w < 8; ++w) acc += part[w];
    atomicAdd(loss_slot, acc);
  }
}

// =====================================================================
// pitch embedding add: x2 = x + embed_pitch[searchsorted_left(pbins, p)]
// also records the energy bin index for fusion into the expand gather
// =====================================================================
__global__ __launch_bounds__(128)
void va_embed_pitch(const float* __restrict__ x, const float* __restrict__ pitches,
                    const float* __restrict__ energies,
                    const float* __restrict__ pbins, const float* __restrict__ ebins,
                    const float* __restrict__ embP, int* __restrict__ eidx,
                    float* __restrict__ x2)
{
  const int row = blockIdx.x;
  const float pv = pitches[row];
  int lo = 0, hi = 255;                      // searchsorted 'left' over 255 bins
  while (lo < hi) { int mid = (lo + hi) >> 1; if (pbins[mid] < pv) lo = mid + 1; else hi = mid; }
  if (threadIdx.x == 0) {
    const float ev = energies[row];
    int l2 = 0, h2 = 255;
    while (l2 < h2) { int mid = (l2 + h2) >> 1; if (ebins[mid] < ev) l2 = mid + 1; else h2 = mid; }
    eidx[row] = l2;
  }
  const float* xr = x + (size_t)row * D_;
  const float* er = embP + (size_t)lo * D_;
  float* o = x2 + (size_t)row * D_;
  for (int c = threadIdx.x; c < D_; c += 128) o[c] = xr[c] + er[c];
}

// =====================================================================
// per-batch inclusive scan of durations; out_lens = cs[T-1]
// =====================================================================
__global__ __launch_bounds__(1024)
void va_scan_durations(const int* __restrict__ dur, int* __restrict__ cs,
                       float* __restrict__ out_lens)
{
  __shared__ int buf[T_];
  const int b = blockIdx.x, t = threadIdx.x;
  buf[t] = dur[b * T_ + t];
  __syncthreads();
  for (int off = 1; off < T_; off <<= 1) {
    int v = (t >= off) ? buf[t - off] : 0;
    __syncthreads();
    buf[t] += v;
    __syncthreads();
  }
  cs[b * T_ + t] = buf[t];
  if (t == T_ - 1) out_lens[b] = (float)buf[t];
}

// =====================================================================
// expand_lengths gather with fused energy-embedding add + validity mask
// one wave per output row; grid: (B*MAXOUT)/8 blocks of 256 threads
// =====================================================================
__global__ __launch_bounds__(256)
void va_expand_gather(const float* __restrict__ x2, const float* __restrict__ embE,
                      const int* __restrict__ eidx, const int* __restrict__ cs,
                      float* __restrict__ out)
{
  const int wid  = blockIdx.x * 8 + (threadIdx.x >> 5);
  const int lane = threadIdx.x & 31;
  const int b   = wid >> 12;           // MAXOUT = 4096 rows per batch
  const int pos = wid & (MAXOUT_ - 1);
  const int* csb = cs + b * T_;
  const int lens = csb[T_ - 1];
  int lo = 0, hi = T_;                 // searchsorted 'right'
  while (lo < hi) { int mid = (lo + hi) >> 1; if (csb[mid] <= pos) lo = mid + 1; else hi = mid; }
  const int idx = lo < (T_ - 1) ? lo : (T_ - 1);
  float* orow = out + (size_t)wid * D_;
  if (pos < lens) {
    const float* xr = x2 + (size_t)(b * T_ + idx) * D_;
    const float* er = embE + (size_t)eidx[b * T_ + idx] * D_;
#pragma unroll
    for (int j = 0; j < 3; ++j) {
      const int c = (j * 32 + lane) * 4;
      float4 xv = *(const float4*)(xr + c);
      float4 ev = *(const float4*)(er + c);
      *(float4*)(orow + c) = make_float4(xv.x + ev.x, xv.y + ev.y, xv.z + ev.z, xv.w + ev.w);
    }
  } else {
#pragma unroll
    for (int j = 0; j < 3; ++j)
      *(float4*)(orow + (j * 32 + lane) * 4) = make_float4(0.f, 0.f, 0.f, 0.f);
  }
}

__global__ void va_init_losses(float* loss3) {
  if (threadIdx.x < 3) loss3[threadIdx.x] = 0.0f;
}

// =====================================================================
extern "C" void kernel_launch(void* const* d_in, const int* in_sizes, int n_in,
                              void* d_out, int out_size, void* d_ws, size_t ws_size,
                              hipStream_t stream)
{
  (void)in_sizes; (void)n_in; (void)out_size; (void)ws_size;
  // setup_inputs order (tuples flattened): x, x_mask, durations, pitches,
  // energies, dp[10], pp[10], ep[10], pitch_bins, embed_pitch, energy_bins,
  // embed_energy.  Predictor tuple: c1w,c1b,l1s,l1b,c2w,c2b,l2s,l2b,pw,pb
  const float* x        = (const float*)d_in[0];
  const int*   dur      = (const int*)  d_in[2];
  const float* pitches  = (const float*)d_in[3];
  const float* energies = (const float*)d_in[4];
  const float* const* dp = (const float* const*)&d_in[5];
  const float* const* pp = (const float* const*)&d_in[15];
  const float* const* ep = (const float* const*)&d_in[25];
  const float* pbins = (const float*)d_in[35];
  const float* embP  = (const float*)d_in[36];
  const float* ebins = (const float*)d_in[37];
  const float* embE  = (const float*)d_in[38];

  float* out = (float*)d_out;
  const size_t XUP = (size_t)B_ * MAXOUT_ * D_;
  float* xup      = out;                 // (B, 4096, 384)
  float* out_lens = out + XUP;           // (B,)
  float* losses   = out + XUP + B_;      // dur, pitch, energy

  char* w = (char*)d_ws;
  const size_t XSZ = (size_t)B_ * T_ * D_ * sizeof(float);
  float* x2   = (float*)w; w += XSZ;     // x + pitch_emb
  float* h1   = (float*)w; w += XSZ;     // layer-1 output (reused per predictor)
  float* h2   = (float*)w; w += XSZ;     // layer-2 output (reused per predictor)
  int*   cs   = (int*)w;   w += (size_t)B_ * T_ * sizeof(int);
  int*   eidx = (int*)w;   w += (size_t)B_ * T_ * sizeof(int);

  va_init_losses<<<1, 32, 0, stream>>>(losses);
  va_embed_pitch<<<B_ * T_, 128, 0, stream>>>(x, pitches, energies, pbins, ebins,
                                              embP, eidx, x2);

  // duration predictor (input x)
  va_conv_ln_wmma<<<2048, 256, 0, stream>>>(x,  dp[0], dp[1], dp[2], dp[3], h1);
  va_conv_ln_wmma<<<2048, 256, 0, stream>>>(h1, dp[4], dp[5], dp[6], dp[7], h2);
  va_proj_loss<<<4096, 256, 0, stream>>>(h2, dp[8], dp[9], pitches, dur, 0, losses + 0);

  // pitch predictor (input x)
  va_conv_ln_wmma<<<2048, 256, 0, stream>>>(x,  pp[0], pp[1], pp[2], pp[3], h1);
  va_conv_ln_wmma<<<2048, 256, 0, stream>>>(h1, pp[4], pp[5], pp[6], pp[7], h2);
  va_proj_loss<<<4096, 256, 0, stream>>>(h2, pp[8], pp[9], pitches, dur, 1, losses + 1);

  // energy predictor (input x2 = x + pitch_emb)
  va_conv_ln_wmma<<<2048, 256, 0, stream>>>(x2, ep[0], ep[1], ep[2], ep[3], h1);
  va_conv_ln_wmma<<<2048, 256, 0, stream>>>(h1, ep[4], ep[5], ep[6], ep[7], h2);
  va_proj_loss<<<4096, 256, 0, stream>>>(h2, ep[8], ep[9], energies, dur, 1, losses + 2);

  // expand_lengths (energy embedding add fused into gather)
  va_scan_durations<<<B_, 1024, 0, stream>>>(dur, cs, out_lens);
  va_expand_gather<<<(B_ * MAXOUT_) / 8, 256, 0, stream>>>(x2, embE, eidx, cs, xup);
}